// BoundaryLoss_63660005261829
// MI455X (gfx1250) — compile-verified
//
#include <hip/hip_runtime.h>
#include <hip/hip_bf16.h>
#include <math.h>

// Problem constants (fixed by reference setup_inputs)
#define NB   2          // batch
#define NC   4          // classes
#define HH   512
#define WW   512
#define HW   (HH * WW)          // 262144
#define NPLANES (NB * NC)       // 8
#define BIGD 1.0e6f

#define COLT 8                  // columns per min-plus tile
#define NTILES (WW / COLT)      // 64
#define K2_BLOCKS (NPLANES * NTILES)   // 512 partial sums

typedef float v2f __attribute__((ext_vector_type(2)));
typedef float v8f __attribute__((ext_vector_type(8)));

// ---------------------------------------------------------------------------
// Kernel 0: zero the class-present flags (ws is poisoned 0xAA; we must init).
// ---------------------------------------------------------------------------
__global__ void k0_init_flags(unsigned* flags) {
    int t = threadIdx.x;
    if (t < NPLANES) flags[t] = 0u;
}

// ---------------------------------------------------------------------------
// Kernel 1: per-row 1-D EDT pass along W for both transforms.
//   forward:  d = mask ? carry+1 : 0   (carry init 1e6)  -> stored as df
//   backward: same from the right; g = min(df, db); store g*g.
// Exactly matches the reference scan (all values are integers < 2^24, so
// float adds/mins are exact). Also records class presence per plane.
// One thread per (plane,row): 4096 threads.
// ---------------------------------------------------------------------------
__global__ void k1_rowpass(const int* __restrict__ tgt,
                           float* __restrict__ g2pos,
                           float* __restrict__ g2neg,
                           unsigned* __restrict__ flags) {
    int gid = blockIdx.x * blockDim.x + threadIdx.x;
    if (gid >= NPLANES * HH) return;
    int plane = gid >> 9;          // /512
    int row   = gid & (HH - 1);
    int n   = plane >> 2;
    int cls = plane & 3;

    const int* trow = tgt + (size_t)n * HW + (size_t)row * WW;
    float* gp = g2pos + (size_t)plane * HW + (size_t)row * WW;
    float* gn = g2neg + (size_t)plane * HW + (size_t)row * WW;

    float cp = BIGD, cn = BIGD;
    bool any = false;
    for (int c = 0; c < WW; ++c) {
        bool oh = (trow[c] == cls);
        any = any || oh;
        cp = oh ? (cp + 1.0f) : 0.0f;   // dist to nearest non-class, from left
        gp[c] = cp;
        cn = (!oh) ? (cn + 1.0f) : 0.0f; // dist to nearest class, from left
        gn[c] = cn;
    }
    cp = BIGD; cn = BIGD;
    for (int c = WW - 1; c >= 0; --c) {
        bool oh = (trow[c] == cls);
        cp = oh ? (cp + 1.0f) : 0.0f;
        float g = fminf(gp[c], cp);
        gp[c] = g * g;
        cn = (!oh) ? (cn + 1.0f) : 0.0f;
        g = fminf(gn[c], cn);
        gn[c] = g * g;
    }
    if (any) atomicOr(&flags[plane], 1u);
}

// ---------------------------------------------------------------------------
// Kernel 2 (hot loop): brute-force min-plus along H on an LDS-resident
// column tile, fused with sqrt/softmax/mask/product and block reduction.
// Grid: 8 planes x 64 column tiles = 512 blocks, 256 threads (8 waves).
// Thread (col = tid&7, rsub = tid>>3) owns rows r = rsub + 32k, k=0..15.
// The (r-rp)^2 offset is computed as exact-integer mul + cvt so the float
// add matches the reference's g2 + off rounding bit-for-bit.
// g2 tiles (16 MB total across grid) were just written: they live in the
// 192 MB L2, so this kernel is VALU-bound, not bandwidth-bound.
// ---------------------------------------------------------------------------
__global__ void __launch_bounds__(256)
k2_minplus_fused(const float* __restrict__ inp,
                 const float* __restrict__ g2pos,
                 const float* __restrict__ g2neg,
                 const unsigned* __restrict__ flags,
                 float* __restrict__ partials) {
    __shared__ float sp[HH * COLT];   // 16 KB
    __shared__ float sn[HH * COLT];   // 16 KB

    int plane = blockIdx.x >> 6;       // / NTILES
    int tile  = blockIdx.x & (NTILES - 1);
    int col0  = tile * COLT;
    int n   = plane >> 2;
    int cls = plane & 3;
    int tid = threadIdx.x;

    // Reference zeroes sdf for class 0 and absent classes -> contribution 0.
    if (cls == 0 || flags[plane] == 0u) {
        if (tid == 0) partials[blockIdx.x] = 0.0f;
        return;
    }

    const float* gpg = g2pos + (size_t)plane * HW;
    const float* gng = g2neg + (size_t)plane * HW;
    for (int i = tid; i < HH * COLT; i += 256) {
        int r = i >> 3;          // / COLT
        int c = i & (COLT - 1);
        sp[i] = gpg[(size_t)r * WW + col0 + c];
        sn[i] = gng[(size_t)r * WW + col0 + c];
    }
    // Warm the softmax logit lines we'll need after the min-plus loop.
    {
        const float* px = inp + (size_t)n * NC * HW + (size_t)(tid >> 3) * WW + col0 + (tid & 7);
        __builtin_prefetch(px, 0, 1);
        __builtin_prefetch(px + HW, 0, 1);
        __builtin_prefetch(px + 2 * HW, 0, 1);
        __builtin_prefetch(px + 3 * HW, 0, 1);
    }
    __syncthreads();

    int col  = tid & (COLT - 1);
    int rsub = tid >> 3;               // 0..31

    float accP[16], accN[16];
#pragma unroll
    for (int k = 0; k < 16; ++k) { accP[k] = 3.402823466e38f; accN[k] = 3.402823466e38f; }

    for (int rp = 0; rp < HH; ++rp) {
        float gp = sp[(rp << 3) | col];
        float gn = sn[(rp << 3) | col];
        int base = rsub - rp;
#pragma unroll
        for (int k = 0; k < 16; ++k) {
            int dr = base + (k << 5);
            float off = (float)(dr * dr);          // exact integer <= 511^2
            accP[k] = fminf(accP[k], gp + off);
            accN[k] = fminf(accN[k], gn + off);
        }
    }

    // sqrt -> signed boundary, softmax prob of this class, accumulate product
    float contrib = 0.0f;
    int colg = col0 + col;
    const float* xb = inp + (size_t)n * NC * HW + colg;
#pragma unroll 4
    for (int k = 0; k < 16; ++k) {
        int r = rsub + (k << 5);
        float sdf = sqrtf(accN[k]) - sqrtf(accP[k]);   // neg_edt - pos_edt
        const float* px = xb + (size_t)r * WW;
        float x0 = px[0];
        float x1 = px[HW];
        float x2 = px[2 * HW];
        float x3 = px[3 * HW];
        float m = fmaxf(fmaxf(x0, x1), fmaxf(x2, x3));
        float e0 = expf(x0 - m);
        float e1 = expf(x1 - m);
        float e2 = expf(x2 - m);
        float e3 = expf(x3 - m);
        float denom = (e0 + e1) + (e2 + e3);
        float ec = (cls == 1) ? e1 : (cls == 2) ? e2 : e3;  // cls>=1 here
        contrib += (ec / denom) * sdf;
    }

    // Block reduction (reuse sp after barrier)
    __syncthreads();
    sp[tid] = contrib;
    __syncthreads();
    for (int s = 128; s > 0; s >>= 1) {
        if (tid < s) sp[tid] += sp[tid + s];
        __syncthreads();
    }
    if (tid == 0) partials[blockIdx.x] = sp[0];
}

// ---------------------------------------------------------------------------
// Kernel 3: WMMA reduction of the 512 partial sums. One wave32, EXEC all-1s.
// A = ones(16x4) so D[i][j] = sum_k B[k][j]; summing row 0 over j gives the
// total of all 64 values packed into B regardless of B's lane packing.
// Accumulates across 8 issues through the C operand.
// ---------------------------------------------------------------------------
__global__ void k3_reduce_wmma(const float* __restrict__ partials,
                               float* __restrict__ out) {
    int lane = threadIdx.x;   // 0..31
    v2f a; a[0] = 1.0f; a[1] = 1.0f;
    v8f c = {};
#pragma unroll
    for (int it = 0; it < K2_BLOCKS / 64; ++it) {
        v2f b;
        b[0] = partials[it * 64 + lane];
        b[1] = partials[it * 64 + 32 + lane];
        c = __builtin_amdgcn_wmma_f32_16x16x4_f32(
                /*neg_a=*/false, a, /*neg_b=*/false, b,
                /*c_mod=*/(short)0, c, /*reuse_a=*/false, /*reuse_b=*/false);
    }
    __shared__ float red[32];
    red[lane] = c[0];                 // VGPR0, lanes 0..15 = D[0][0..15]
    __syncthreads();
    if (lane == 0) {
        float s = 0.0f;
        for (int j = 0; j < 16; ++j) s += red[j];
        out[0] = s * (1.0f / (float)(NB * NC * HW));   // mean over N*C*H*W
    }
}

// ---------------------------------------------------------------------------
extern "C" void kernel_launch(void* const* d_in, const int* in_sizes, int n_in,
                              void* d_out, int out_size, void* d_ws, size_t ws_size,
                              hipStream_t stream) {
    const float* inputs = (const float*)d_in[0];   // [2,4,512,512] f32
    const int*   tgt    = (const int*)d_in[1];     // [2,512,512] int

    // Workspace layout: g2pos(8MB) | g2neg(8MB) | partials(2KB) | flags(32B)
    float*    g2pos    = (float*)d_ws;
    float*    g2neg    = g2pos + (size_t)NPLANES * HW;
    float*    partials = g2neg + (size_t)NPLANES * HW;
    unsigned* flags    = (unsigned*)(partials + K2_BLOCKS);

    k0_init_flags<<<1, 32, 0, stream>>>(flags);

    k1_rowpass<<<(NPLANES * HH + 255) / 256, 256, 0, stream>>>(
        tgt, g2pos, g2neg, flags);

    k2_minplus_fused<<<K2_BLOCKS, 256, 0, stream>>>(
        inputs, g2pos, g2neg, flags, partials);

    k3_reduce_wmma<<<1, 32, 0, stream>>>(partials, (float*)d_out);
}